// StreamingKVCache_87857851007393
// MI455X (gfx1250) — compile-verified
//
#include <hip/hip_runtime.h>
#include <cstdint>

// StreamingKVCache.evict_kv for MI455X (gfx1250).
// Pure HBM-bound row gather: 256MB read + 256MB write ~= 22us at 23.3 TB/s.
// Uses the CDNA5 async global<->LDS DMA path (ASYNCcnt) with b128 transfers.

#define BSZ   8
#define TT    4096
#define HH    8
#define DD    128
#define PG    16
#define ROW   (HH * DD)            // 1024 floats per token per K (and per V)
#define PAGEF (2 * PG * ROW)       // 32768 floats per page
#define VOFF  (PG * ROW)           // 16384 floats: V block offset inside a page

__global__ __launch_bounds__(256)
void evict_kv_kernel(const float* __restrict__ kv,
                     const int*   __restrict__ num_evicts,
                     const int*   __restrict__ cachelens,
                     const int*   __restrict__ sink_p,
                     float*       __restrict__ out) {
    // 8 KB staging tile: K row in [0,1024), V row in [1024,2048) floats.
    __shared__ float smem[2 * ROW];

    const int ft  = blockIdx.x;          // destination flat token index
    const int b   = ft >> 12;            // / 4096
    const int pos = ft & (TT - 1);

    const int ev     = num_evicts[b];
    const int target = cachelens[b] - ev;
    const int sink   = sink_p[0];

    int src = pos;
    if (pos >= sink && pos < target) src = pos + ev;
    if (src > TT - 1) src = TT - 1;

    const int sft = (b << 12) + src;     // source flat token index

    // Row base offsets (floats). Each K/V row is 1024 contiguous floats.
    const long kSrc = (long)(sft >> 4) * PAGEF + (long)(sft & 15) * ROW;
    const long kDst = (long)(ft  >> 4) * PAGEF + (long)(ft  & 15) * ROW;

    const int tid = threadIdx.x;         // 0..255, one float4 each for K and V

    const float* gK = kv + kSrc +        tid * 4;
    const float* gV = kv + kSrc + VOFF + tid * 4;
    float*       oK = out + kDst +        tid * 4;
    float*       oV = out + kDst + VOFF + tid * 4;

    // LDS byte offsets (generic shared pointer low 32 bits == LDS offset).
    const uint32_t ldsK = (uint32_t)(uintptr_t)(&smem[tid * 4]);
    const uint32_t ldsV = (uint32_t)(uintptr_t)(&smem[ROW + tid * 4]);

    // Async DMA: global -> LDS (b128 per lane), tracked by ASYNCcnt.
    asm volatile("global_load_async_to_lds_b128 %0, %1, off"
                 :: "v"(ldsK), "v"((uint64_t)(uintptr_t)gK) : "memory");
    asm volatile("global_load_async_to_lds_b128 %0, %1, off"
                 :: "v"(ldsV), "v"((uint64_t)(uintptr_t)gV) : "memory");

    // Each lane stores exactly the LDS region it loaded, so a per-wave
    // asynccnt drain is sufficient (no workgroup barrier needed).
    asm volatile("s_wait_asynccnt 0" ::: "memory");

    // Async DMA: LDS -> global (b128 per lane).
    asm volatile("global_store_async_from_lds_b128 %0, %1, off"
                 :: "v"((uint64_t)(uintptr_t)oK), "v"(ldsK) : "memory");
    asm volatile("global_store_async_from_lds_b128 %0, %1, off"
                 :: "v"((uint64_t)(uintptr_t)oV), "v"(ldsV) : "memory");

    asm volatile("s_wait_asynccnt 0" ::: "memory");
}

extern "C" void kernel_launch(void* const* d_in, const int* in_sizes, int n_in,
                              void* d_out, int out_size, void* d_ws, size_t ws_size,
                              hipStream_t stream) {
    // setup_inputs order: kv_cache(f32), num_evicts(i32), cachelens(i32),
    // num_sink_tokens(i32 scalar), n_local_heads, head_dim, page_size.
    const float* kv   = (const float*)d_in[0];
    const int*   ev   = (const int*)  d_in[1];
    const int*   cl   = (const int*)  d_in[2];
    const int*   sink = (const int*)  d_in[3];
    float*       out  = (float*)d_out;

    // One block per destination token row (K+V = 8 KB moved per block).
    dim3 grid(BSZ * TT);
    dim3 block(256);
    evict_kv_kernel<<<grid, block, 0, stream>>>(kv, ev, cl, sink, out);
}